// PointGRN_55868934586530
// MI455X (gfx1250) — compile-verified
//
#include <hip/hip_runtime.h>
#include <math.h>

// ---------------------------------------------------------------------------
// PointGRN for MI455X (gfx1250, wave32).
// Pass 1: per-segment per-channel sum-of-squares via V_WMMA_F32_16X16X4_F32
//         (reduction-as-matmul: A = squared feats (16ch x 4pts), B = ones).
//         Segment logic hoisted out of the hot loop; 32-bit indexing only.
// Pass 2: tiny per-segment normalization -> mult[b,c] = 1 + gamma*rn.
// Pass 3: streaming out = feat*mult[batch] + beta with NT load/store and
//         incremental segment tracking (monotone point index per thread).
// ---------------------------------------------------------------------------

typedef float v2f __attribute__((ext_vector_type(2)));
typedef float v4f __attribute__((ext_vector_type(4)));
typedef float v8f __attribute__((ext_vector_type(8)));

#define GRN_EPS 1e-6f

__device__ __forceinline__ int seg_of_s(int p, const int* soff, int B) {
  // jnp.searchsorted(offset, p, side="right") == count(offset[j] <= p)
  int b = 0;
  for (int j = 0; j < B; ++j) b += (p >= soff[j]) ? 1 : 0;
  return (b < B - 1) ? b : (B - 1);
}

__global__ void grn_zero_kernel(float* p, int n) {
  for (int i = threadIdx.x + blockIdx.x * blockDim.x; i < n;
       i += blockDim.x * gridDim.x)
    p[i] = 0.0f;
}

__device__ __forceinline__ v8f grn_wmma_acc(v2f a, v2f bones, v8f acc) {
  return __builtin_amdgcn_wmma_f32_16x16x4_f32(
      /*neg_a=*/false, a, /*neg_b=*/false, bones,
      /*c_mod=*/(short)0, acc, /*reuse_a=*/false, /*reuse_b=*/false);
}

// Flush WMMA accumulator: D[m][n] is identical across n (B was all-ones), and
// per the 16x16 f32 C/D layout lane 0 holds rows 0..7 (VGPR j = row j) and
// lane 16 holds rows 8..15.
__device__ __forceinline__ void grn_flush64(const v8f acc, int b, int base,
                                            int lane, float* sumsq) {
  constexpr int C = 64;
  if (lane == 0) {
#pragma unroll
    for (int j = 0; j < 8; ++j)
      atomicAdd(&sumsq[b * C + base + j], acc[j]);
  } else if (lane == 16) {
#pragma unroll
    for (int j = 0; j < 8; ++j)
      atomicAdd(&sumsq[b * C + base + 8 + j], acc[j]);
  }
}

// Pass 1 (C == 64): each wave owns 16 channels (base = 16*(wave&3)) of a
// contiguous run of 4-point groups. A-matrix (16x4 f32, 2 VGPRs): lanes 0-15
// carry points p0,p1 / lanes 16-31 carry p2,p3 for channel base+(lane&15).
__global__ void grn_sumsq_wmma_kernel(const float* __restrict__ feat,
                                      const int* __restrict__ offset,
                                      float* __restrict__ sumsq,
                                      int N, int B) {
  constexpr int C = 64;
  __shared__ int soff[64];
  if (threadIdx.x < B && threadIdx.x < 64) soff[threadIdx.x] = offset[threadIdx.x];
  __syncthreads();

  const int lane       = threadIdx.x & 31;
  const int wave       = threadIdx.x >> 5;         // 0..7
  const int base       = (wave & 3) * 16;          // channel base
  const int rowInBlk   = wave >> 2;                // 0..1
  const int rowsPerBlk = (blockDim.x >> 5) >> 2;   // 2 for 256 threads

  const int totalGroups = N >> 2;                  // groups of 4 points
  const int numRows = gridDim.x * rowsPerBlk;
  const int R = blockIdx.x * rowsPerBlk + rowInBlk;
  const int gpr = (totalGroups + numRows - 1) / numRows;
  const int g0 = R * gpr;
  const int g1 = min(g0 + gpr, totalGroups);

  const int half = lane >> 4;            // 0 -> points p0,p1 ; 1 -> p2,p3
  const int ch   = base + (lane & 15);

  v2f bones; bones.x = 1.0f; bones.y = 1.0f;       // B = ones(4x16)

  int g = g0;
  while (g < g1) {
    const int pf = g << 2;
    // segment of the group's first point (wave-uniform) + its end point
    const int b = seg_of_s(pf, soff, B);
    const int segEnd = soff[b];                    // first point NOT in seg b
    const int ge = min(g1, segEnd >> 2);           // groups fully inside seg b

    if (ge > g) {
      // ---- HOT LOOP: no segment logic, two independent WMMA chains ----
      v8f acc0 = {};
      v8f acc1 = {};
      const float* ptr = feat + ((g << 2) + 2 * half) * C + ch;
      int i = g;
      for (; i + 1 < ge; i += 2) {
        const float x0 = ptr[0];
        const float x1 = ptr[C];
        const float y0 = ptr[4 * C];
        const float y1 = ptr[5 * C];
        v2f a0; a0.x = x0 * x0; a0.y = x1 * x1;
        v2f a1; a1.x = y0 * y0; a1.y = y1 * y1;
        acc0 = grn_wmma_acc(a0, bones, acc0);
        acc1 = grn_wmma_acc(a1, bones, acc1);
        ptr += 8 * C;
      }
      if (i < ge) {
        const float x0 = ptr[0];
        const float x1 = ptr[C];
        v2f a0; a0.x = x0 * x0; a0.y = x1 * x1;
        acc0 = grn_wmma_acc(a0, bones, acc0);
      }
      acc0 = acc0 + acc1;
      grn_flush64(acc0, b, base, lane, sumsq);
      g = ge;
    }

    // rare: group straddles a segment boundary (segEnd % 4 != 0)
    if (g < g1 && (g << 2) < segEnd) {
      const int pA = (g << 2) + 2 * half;
      const float x0 = feat[pA * C + ch];
      const float x1 = feat[(pA + 1) * C + ch];
      const int b0 = seg_of_s(pA, soff, B);
      const int b1 = seg_of_s(pA + 1, soff, B);
      atomicAdd(&sumsq[b0 * C + ch], x0 * x0);
      atomicAdd(&sumsq[b1 * C + ch], x1 * x1);
      ++g;
    }
  }

  // tail points (N % 4), handled by block 0 / wave 0
  if (blockIdx.x == 0 && wave == 0) {
    for (int p = totalGroups << 2; p < N; ++p) {
      const int b = seg_of_s(p, soff, B);
      for (int cc = lane; cc < C; cc += 32) {
        const float v = feat[p * C + cc];
        atomicAdd(&sumsq[b * C + cc], v * v);
      }
    }
  }
}

// Generic scalar fallback for pass 1 (used if C != 64 or B > 64).
__global__ void grn_sumsq_scalar_kernel(const float* __restrict__ feat,
                                        const int* __restrict__ offset,
                                        float* __restrict__ sumsq,
                                        int N, int C, int B) {
  __shared__ int soff[64];
  if (threadIdx.x < B && threadIdx.x < 64) soff[threadIdx.x] = offset[threadIdx.x];
  __syncthreads();
  const int total = N * C;
  const int stride = blockDim.x * gridDim.x;
  for (int e = threadIdx.x + blockIdx.x * blockDim.x; e < total; e += stride) {
    const int p = e / C;
    const int c = e - p * C;
    const int b = seg_of_s(p, soff, B);
    const float v = feat[e];
    atomicAdd(&sumsq[b * C + c], v * v);
  }
}

// Pass 2: response = sqrt(sumsq); mult[b,c] = 1 + gamma[c] * r / (mean_c r + eps)
__global__ void grn_scale_kernel(const float* __restrict__ sumsq,
                                 const float* __restrict__ gamma,
                                 float* __restrict__ mult, int C, int B) {
  __shared__ float red[1024];
  __shared__ float s_mean;
  const int c = threadIdx.x;
  for (int b = 0; b < B; ++b) {
    float r = 0.0f;
    if (c < C) {
      r = sqrtf(sumsq[b * C + c]);
      red[c] = r;
    }
    __syncthreads();
    if (c == 0) {
      float s = 0.0f;
      for (int j = 0; j < C; ++j) s += red[j];
      s_mean = s / (float)C;
    }
    __syncthreads();
    if (c < C) mult[b * C + c] = 1.0f + gamma[c] * (r / (s_mean + GRN_EPS));
    __syncthreads();
  }
}

// Pass 3 (C power of two, B*C <= 2048): streaming float4 apply, NT hints,
// incremental segment tracking (p is monotone per thread in grid-stride).
__global__ void grn_apply_kernel(const float* __restrict__ feat,
                                 const int* __restrict__ offset,
                                 const float* __restrict__ mult,
                                 const float* __restrict__ beta,
                                 float* __restrict__ out,
                                 int N, int B, int cShift) {
  const int C = 1 << cShift;
  __shared__ __align__(16) float s_mult[2048];
  __shared__ __align__(16) float s_beta[256];
  __shared__ int soff[64];
  for (int i = threadIdx.x; i < B * C; i += blockDim.x) s_mult[i] = mult[i];
  for (int i = threadIdx.x; i < C; i += blockDim.x) s_beta[i] = beta[i];
  if (threadIdx.x < B && threadIdx.x < 64) soff[threadIdx.x] = offset[threadIdx.x];
  __syncthreads();

  const int total4 = (N << cShift) >> 2;           // N*C/4 (< 2^31)
  const int stride = blockDim.x * gridDim.x;

  int b = 0;
  int nb = soff[0];                                // next boundary point
  for (int t = threadIdx.x + blockIdx.x * blockDim.x; t < total4; t += stride) {
    const int e = t << 2;
    const int p = e >> cShift;                     // point index (monotone)
    const int c = e & (C - 1);                     // multiple of 4
    while (b < B - 1 && p >= nb) { ++b; nb = soff[b]; }
    // last use of feat -> non-temporal (hits L2 from pass 1, then discard)
    const v4f f = __builtin_nontemporal_load((const v4f*)feat + t);
    const v4f m = *(const v4f*)&s_mult[(b << cShift) + c];
    const v4f bb = *(const v4f*)&s_beta[c];
    v4f o;
    o.x = fmaf(f.x, m.x, bb.x);
    o.y = fmaf(f.y, m.y, bb.y);
    o.z = fmaf(f.z, m.z, bb.z);
    o.w = fmaf(f.w, m.w, bb.w);
    __builtin_nontemporal_store(o, (v4f*)out + t); // streaming output
  }
}

// Generic scalar apply fallback.
__global__ void grn_apply_scalar_kernel(const float* __restrict__ feat,
                                        const int* __restrict__ offset,
                                        const float* __restrict__ mult,
                                        const float* __restrict__ beta,
                                        float* __restrict__ out,
                                        int N, int C, int B) {
  __shared__ int soff[64];
  if (threadIdx.x < B && threadIdx.x < 64) soff[threadIdx.x] = offset[threadIdx.x];
  __syncthreads();
  const int total = N * C;
  const int stride = blockDim.x * gridDim.x;
  for (int e = threadIdx.x + blockIdx.x * blockDim.x; e < total; e += stride) {
    const int p = e / C;
    const int c = e - p * C;
    const int b = seg_of_s(p, soff, B);
    out[e] = fmaf(feat[e], mult[b * C + c], beta[c]);
  }
}

extern "C" void kernel_launch(void* const* d_in, const int* in_sizes, int n_in,
                              void* d_out, int out_size, void* d_ws, size_t ws_size,
                              hipStream_t stream) {
  const float* feat   = (const float*)d_in[0];
  const int*   offset = (const int*)d_in[1];
  const float* gamma  = (const float*)d_in[2];
  const float* beta   = (const float*)d_in[3];
  float* out = (float*)d_out;

  const int C = in_sizes[2];                 // gamma has C elements
  const int B = in_sizes[1];                 // offset has B elements
  const int N = in_sizes[0] / C;

  float* sumsq = (float*)d_ws;               // B*C floats
  float* mult  = sumsq + (size_t)B * C;      // B*C floats

  grn_zero_kernel<<<1, 256, 0, stream>>>(sumsq, B * C);

  if (C == 64 && B <= 64) {
    grn_sumsq_wmma_kernel<<<512, 256, 0, stream>>>(feat, offset, sumsq, N, B);
  } else {
    grn_sumsq_scalar_kernel<<<1024, 256, 0, stream>>>(feat, offset, sumsq, N, C, B);
  }

  const int scaleThreads = (C <= 1024) ? ((C + 63) / 64) * 64 : 1024;
  grn_scale_kernel<<<1, scaleThreads, 0, stream>>>(sumsq, gamma, mult, C, B);

  // C power of two?
  int cShift = -1;
  for (int s = 0; s < 12; ++s)
    if ((1 << s) == C) cShift = s;

  if (cShift >= 0 && (B * C <= 2048) && (C <= 256) && (B <= 64) && C >= 4) {
    grn_apply_kernel<<<2048, 256, 0, stream>>>(feat, offset, mult, beta, out,
                                               N, B, cShift);
  } else {
    grn_apply_scalar_kernel<<<2048, 256, 0, stream>>>(feat, offset, mult, beta,
                                                      out, N, C, B);
  }
}